// ChamferDistance_15693810500250
// MI455X (gfx1250) — compile-verified
//
#include <hip/hip_runtime.h>
#include <math.h>

typedef float v2f __attribute__((ext_vector_type(2)));
typedef float v8f __attribute__((ext_vector_type(8)));

// ---------------------------------------------------------------------------
// Chamfer distance via V_WMMA_F32_16X16X4_F32 (fp32, K=4 >= D=3, no precision
// loss).  Distance identity folded entirely into the matrix op:
//   A(16x4)  = [-2p0, -2p1, -2p2, 1]     (M = lane&15, K pair split by half)
//   B(4x16)  = [ q0 ,  q1 ,  q2 , |q|^2] (N = lane&15, K pair split by half)
//   C(16x16) = |p|^2 per row
// => D[n][m] = ||p_n - q_m||^2 straight out of the WMMA; sqrt is monotonic so
// we min over squared distances and sqrt once per row.
// Chamfer is symmetric: run the same row-min kernel twice with P/Q swapped,
// so no column-min / atomic-min pass is ever needed.
// ---------------------------------------------------------------------------

// Prologue: pack the B-matrix operand registers for every (batch, tile, lane).
// Shared by all 256 row-strip waves of a batch -> computed once, read many.
__global__ __launch_bounds__(256)
void chamfer_pack_kernel(const float* __restrict__ Q, float2* __restrict__ packQ,
                         int Nq, int tilesPerBatch)
{
    const int gid   = blockIdx.x * blockDim.x + threadIdx.x;  // over B*tiles*32
    const int lane  = gid & 31;
    const int tileG = gid >> 5;                // b*tiles + t
    const int b     = tileG / tilesPerBatch;
    const int t     = tileG % tilesPerBatch;
    const int half  = lane >> 4;
    const int l     = lane & 15;

    const float* __restrict__ Qb = Q + (size_t)b * Nq * 3;
    const int base = (t * 16 + l) * 3;
    const float q0 = Qb[base], q1 = Qb[base + 1], q2 = Qb[base + 2];
    const float qsq = fmaf(q0, q0, fmaf(q1, q1, q2 * q2));

    float2 r;
    r.x = half ? q2  : q0;     // K = 2*half
    r.y = half ? qsq : q1;     // K = 2*half+1  (K=3 pad slot carries |q|^2)
    packQ[gid] = r;
}

template <bool PACKED>
__global__ __launch_bounds__(256)
void chamfer_dir_kernel(const float* __restrict__ P, const float* __restrict__ Q,
                        const float2* __restrict__ packQ,
                        int Np, int Nq, int stripsPerBatch,
                        float* __restrict__ partialOut, float* __restrict__ atomicOut,
                        float scale)
{
    __shared__ float lds[8];

    const int tid  = threadIdx.x;
    const int wave = tid >> 5;         // 0..7
    const int lane = tid & 31;
    const int half = lane >> 4;        // selects K pair {0,1} vs {2,3}
    const int l    = lane & 15;

    const int b     = blockIdx.x / stripsPerBatch;
    const int strip = blockIdx.x % stripsPerBatch;
    const int n0    = strip * 128 + wave * 16;   // this wave's 16 P-rows

    const float* __restrict__ Pb = P + (size_t)b * Np * 3;
    const float* __restrict__ Qb = Q + (size_t)b * Nq * 3;

    // ---- A registers: row = n0 + l, K = {2*half, 2*half+1}
    const int pbase = (n0 + l) * 3;
    const float p02 = Pb[pbase + 2 * half];
    const float p1  = Pb[pbase + 1];
    v2f a;
    a.x = -2.0f * p02;
    a.y = half ? 1.0f : (-2.0f * p1);

    // ---- C seed: |p|^2 for row = n0 + v + 8*half
    v8f cseed;
    #pragma unroll
    for (int v = 0; v < 8; ++v) {
        const int r = (n0 + v + 8 * half) * 3;
        const float x0 = Pb[r], x1 = Pb[r + 1], x2 = Pb[r + 2];
        cseed[v] = fmaf(x0, x0, fmaf(x1, x1, x2 * x2));
    }

    v8f m;
    #pragma unroll
    for (int v = 0; v < 8; ++v) m[v] = 3.402823466e38f;

    const int tiles = Nq >> 4;                       // multiple of 4 by design
    const float2* __restrict__ pq =
        PACKED ? (packQ + (size_t)b * tiles * 32 + lane) : nullptr;

    for (int t = 0; t < tiles; t += 4) {
        #pragma unroll
        for (int u = 0; u < 4; ++u) {
            v2f bb;
            if (PACKED) {
                // one b64 load per tile; tile operand fully precomputed
                const float2 bv = pq[(size_t)(t + u) * 32];
                bb.x = bv.x;
                bb.y = bv.y;
            } else {
                // lane-local |q|^2: no cross-lane shuffle in the hot loop
                const int qbase = ((t + u) * 16 + l) * 3;
                const float q0 = Qb[qbase], q1 = Qb[qbase + 1], q2 = Qb[qbase + 2];
                const float qsq = fmaf(q0, q0, fmaf(q1, q1, q2 * q2));
                bb.x = half ? q2  : q0;
                bb.y = half ? qsq : q1;
            }

            // D = A*B + C = squared-distance tile
            const v8f d = __builtin_amdgcn_wmma_f32_16x16x4_f32(
                              false, a, false, bb, (short)0, cseed, false, false);

            #pragma unroll
            for (int v = 0; v < 8; ++v) m[v] = fminf(m[v], d[v]);
        }
    }

    // ---- row-min across the 16 column-lanes of each half (epilogue only)
    #pragma unroll
    for (int off = 1; off < 16; off <<= 1) {
        #pragma unroll
        for (int v = 0; v < 8; ++v)
            m[v] = fminf(m[v], __shfl_xor(m[v], off, 32));
    }

    // ---- sum sqrt(rowmin) over this wave's 16 rows
    float s = 0.0f;
    #pragma unroll
    for (int v = 0; v < 8; ++v) s += sqrtf(fmaxf(m[v], 0.0f));
    s += __shfl_xor(s, 16, 32);

    if (lane == 0) lds[wave] = s;
    __syncthreads();
    if (tid == 0) {
        float bs = 0.0f;
        #pragma unroll
        for (int w = 0; w < 8; ++w) bs += lds[w];
        if (partialOut) partialOut[blockIdx.x] = bs;          // deterministic
        else            atomicAdd(atomicOut + b, bs * scale); // fallback
    }
}

__global__ void chamfer_reduce_kernel(const float* __restrict__ partial,
                                      float* __restrict__ out,
                                      int nb1, int nb2, float inv2N, float inv2M, int Bn)
{
    const int b = threadIdx.x;
    if (b >= Bn) return;
    float s1 = 0.0f;
    for (int k = 0; k < nb1; ++k) s1 += partial[b * nb1 + k];
    const float* p2 = partial + Bn * nb1;
    float s2 = 0.0f;
    for (int k = 0; k < nb2; ++k) s2 += p2[b * nb2 + k];
    out[b] = s1 * inv2N + s2 * inv2M;
}

__global__ void chamfer_zero_kernel(float* __restrict__ p, int n)
{
    const int i = blockIdx.x * blockDim.x + threadIdx.x;
    if (i < n) p[i] = 0.0f;
}

extern "C" void kernel_launch(void* const* d_in, const int* in_sizes, int n_in,
                              void* d_out, int out_size, void* d_ws, size_t ws_size,
                              hipStream_t stream)
{
    const float* x = (const float*)d_in[0];   // [B, N, 3]
    const float* y = (const float*)d_in[1];   // [B, M, 3]
    float* out = (float*)d_out;               // [B]

    const int Bn = 16, N = 4096, M = 4096;
    const int nb1 = N / 128, nb2 = M / 128;         // blocks per batch per dir
    const int tiles1 = M / 16, tiles2 = N / 16;     // column tiles per dir

    const size_t needPart = (size_t)(Bn * (nb1 + nb2)) * sizeof(float);
    const size_t packElems = (size_t)Bn * (tiles1 + tiles2) * 32;
    const size_t needFull = needPart + packElems * sizeof(float2);

    if (d_ws && ws_size >= needFull) {
        float*  partial = (float*)d_ws;
        float2* packY   = (float2*)((char*)d_ws + needPart);
        float2* packX   = packY + (size_t)Bn * tiles1 * 32;

        chamfer_pack_kernel<<<Bn * tiles1 * 32 / 256, 256, 0, stream>>>(
            y, packY, M, tiles1);
        chamfer_pack_kernel<<<Bn * tiles2 * 32 / 256, 256, 0, stream>>>(
            x, packX, N, tiles2);

        chamfer_dir_kernel<true><<<Bn * nb1, 256, 0, stream>>>(
            x, y, packY, N, M, nb1, partial, nullptr, 0.0f);
        chamfer_dir_kernel<true><<<Bn * nb2, 256, 0, stream>>>(
            y, x, packX, M, N, nb2, partial + Bn * nb1, nullptr, 0.0f);

        chamfer_reduce_kernel<<<1, 32, 0, stream>>>(
            partial, out, nb1, nb2, 0.5f / N, 0.5f / M, Bn);
    } else if (d_ws && ws_size >= needPart) {
        float* partial = (float*)d_ws;
        chamfer_dir_kernel<false><<<Bn * nb1, 256, 0, stream>>>(
            x, y, nullptr, N, M, nb1, partial, nullptr, 0.0f);
        chamfer_dir_kernel<false><<<Bn * nb2, 256, 0, stream>>>(
            y, x, nullptr, M, N, nb2, partial + Bn * nb1, nullptr, 0.0f);
        chamfer_reduce_kernel<<<1, 32, 0, stream>>>(
            partial, out, nb1, nb2, 0.5f / N, 0.5f / M, Bn);
    } else {
        chamfer_zero_kernel<<<1, 32, 0, stream>>>(out, Bn);
        chamfer_dir_kernel<false><<<Bn * nb1, 256, 0, stream>>>(
            x, y, nullptr, N, M, nb1, nullptr, out, 0.5f / N);
        chamfer_dir_kernel<false><<<Bn * nb2, 256, 0, stream>>>(
            y, x, nullptr, M, N, nb2, nullptr, out, 0.5f / M);
    }
}